// LiquidOperator_33732673143253
// MI455X (gfx1250) — compile-verified
//
#include <hip/hip_runtime.h>
#include <math.h>

typedef __attribute__((ext_vector_type(2))) float v2f;
typedef __attribute__((ext_vector_type(8))) float v8f;

#define VAR_N   4
#define LEVELS  17
#define NCELLS  56
#define PRED_N  12
#define DDIM    68
#define T_OBS   4096
#define TOTAL_T (T_OBS + PRED_N)   /* 4108 */
#define DT_STEP 0.1f

// ---------------------------------------------------------------------------
// WMMA fp32 GEMM:  C[M,N] = act( A[M,K] @ W[K,N] + bias[N] )
// One wave computes a 64x16 output macro-tile (MT=4 stacked 16x16 tiles)
// with V_WMMA_F32_16X16X4_F32, stepping K by 4. The B fragment is loaded
// once per K-chunk and reused by 4 independent WMMA chains (ILP hides the
// WMMA RAW hazard). All edge handling is branchless: clamped addresses +
// v_cndmask selects, so the hot loop is pure load/valu/wmma.
//
// Fragment layouts (CDNA5 ISA 7.12.2, 32-bit operands):
//   A (16x4):  lanes 0-15 -> M=lane, K={k0,k0+1}; lanes 16-31 -> K={k0+2,k0+3}
//   B (4x16):  lanes 0-15 -> N=lane, K={k0,k0+1}; lanes 16-31 -> K={k0+2,k0+3}
//   C (16x16): VGPR r: lanes 0-15 -> M=r, N=lane; lanes 16-31 -> M=8+r
// ---------------------------------------------------------------------------
#define MT 4

template<bool RELU>
__global__ __launch_bounds__(32)
void wmma_gemm_bias(const float* __restrict__ A, const float* __restrict__ W,
                    const float* __restrict__ bias, float* __restrict__ C,
                    int M, int N, int K)
{
    const int lane = threadIdx.x;            // 0..31
    const int row0 = blockIdx.x * (16 * MT);
    const int col0 = blockIdx.y * 16;

    const int mr  = lane & 15;
    const int kph = (lane >> 4) << 1;        // 0 or 2

    // B column handled by this lane (clamped, masked)
    const int  bcol   = col0 + mr;
    const bool bval   = bcol < N;
    const int  bcolc  = bval ? bcol : (N - 1);

    // A rows handled by this lane (clamped, masked)
    int  arowc[MT];
    bool aval[MT];
    #pragma unroll
    for (int mt = 0; mt < MT; ++mt) {
        const int r = row0 + mt * 16 + mr;
        aval[mt]  = r < M;
        arowc[mt] = aval[mt] ? r : (M - 1);
    }

    v8f acc[MT] = {};
    const int K4 = K & ~3;

    #pragma unroll 2
    for (int k0 = 0; k0 < K4; k0 += 4) {
        const int ka = k0 + kph;
        v2f b;
        {
            const float b0 = W[(size_t)ka * N + bcolc];
            const float b1 = W[(size_t)(ka + 1) * N + bcolc];
            b.x = bval ? b0 : 0.f;
            b.y = bval ? b1 : 0.f;
        }
        v2f a[MT];
        #pragma unroll
        for (int mt = 0; mt < MT; ++mt) {
            const float a0 = A[(size_t)arowc[mt] * K + ka];
            const float a1 = A[(size_t)arowc[mt] * K + ka + 1];
            a[mt].x = aval[mt] ? a0 : 0.f;
            a[mt].y = aval[mt] ? a1 : 0.f;
        }
        #pragma unroll
        for (int mt = 0; mt < MT; ++mt)
            acc[mt] = __builtin_amdgcn_wmma_f32_16x16x4_f32(
                false, a[mt], false, b, (short)0, acc[mt], false, false);
    }

    if (K4 < K) {                            // peeled masked tail (K % 4 != 0)
        const int  ka   = K4 + kph;
        const bool k0v  = ka < K;
        const bool k1v  = (ka + 1) < K;
        const int  kac0 = k0v ? ka : (K - 1);
        const int  kac1 = k1v ? (ka + 1) : (K - 1);
        v2f b;
        {
            const float b0 = W[(size_t)kac0 * N + bcolc];
            const float b1 = W[(size_t)kac1 * N + bcolc];
            b.x = (bval && k0v) ? b0 : 0.f;
            b.y = (bval && k1v) ? b1 : 0.f;
        }
        v2f a[MT];
        #pragma unroll
        for (int mt = 0; mt < MT; ++mt) {
            const float a0 = A[(size_t)arowc[mt] * K + kac0];
            const float a1 = A[(size_t)arowc[mt] * K + kac1];
            a[mt].x = (aval[mt] && k0v) ? a0 : 0.f;
            a[mt].y = (aval[mt] && k1v) ? a1 : 0.f;
        }
        #pragma unroll
        for (int mt = 0; mt < MT; ++mt)
            acc[mt] = __builtin_amdgcn_wmma_f32_16x16x4_f32(
                false, a[mt], false, b, (short)0, acc[mt], false, false);
    }

    // Store with bias + optional ReLU
    const int col   = col0 + mr;
    const int rbase = (lane >> 4) << 3;      // 0 or 8
    if (col < N) {
        const float bv = bias[col];
        #pragma unroll
        for (int mt = 0; mt < MT; ++mt) {
            #pragma unroll
            for (int r = 0; r < 8; ++r) {
                const int row = row0 + mt * 16 + rbase + r;
                if (row < M) {
                    float v = acc[mt][r] + bv;
                    if (RELU) v = v > 0.f ? v : 0.f;
                    C[(size_t)row * N + col] = v;
                }
            }
        }
    }
}

// ---------------------------------------------------------------------------
// LTC recurrence. One block per var (4 blocks, 64 threads). Thread j owns
// cell j; all weights resident in LDS (~21 KB of the 320 KB/WGP); 2 barriers
// per step. Serial critical path: 4108 dependent steps. Writes outputs
// directly in transposed (TOTAL_T, 4, 17) layout so the downstream encoders
// read contiguously.
// ---------------------------------------------------------------------------
__global__ __launch_bounds__(64)
void ltc_scan(const float* __restrict__ pre,    // (4096,68) flat == (4,4096,17)
              const float* __restrict__ wx,     // (4,17,56)
              const float* __restrict__ wh,     // (4,56,56)
              const float* __restrict__ bvec,   // (4,56)
              const float* __restrict__ avec,   // (4,56)
              const float* __restrict__ tau,    // (4,56)
              const float* __restrict__ wout,   // (4,56,17)
              const float* __restrict__ bout,   // (4,17)
              float* __restrict__ vs)           // (4108,4,17)
{
    __shared__ float s_wx[LEVELS * NCELLS];
    __shared__ float s_wh[NCELLS * NCELLS];
    __shared__ float s_wout[NCELLS * LEVELS];
    __shared__ float s_b[NCELLS], s_a[NCELLS], s_it[NCELLS], s_bout[LEVELS];
    __shared__ float s_h[2][NCELLS];
    __shared__ float s_u[LEVELS], s_o[LEVELS];

    const int v   = blockIdx.x;
    const int tid = threadIdx.x;

    for (int i = tid; i < LEVELS * NCELLS; i += blockDim.x)
        s_wx[i] = wx[v * LEVELS * NCELLS + i];
    for (int i = tid; i < NCELLS * NCELLS; i += blockDim.x)
        s_wh[i] = wh[v * NCELLS * NCELLS + i];
    for (int i = tid; i < NCELLS * LEVELS; i += blockDim.x)
        s_wout[i] = wout[v * NCELLS * LEVELS + i];
    if (tid < NCELLS) {
        s_b[tid]    = bvec[v * NCELLS + tid];
        s_a[tid]    = avec[v * NCELLS + tid];
        s_it[tid]   = 1.0f / (tau[v * NCELLS + tid] + 0.5f);
        s_h[0][tid] = 0.0f;
    }
    if (tid < LEVELS) s_bout[tid] = bout[v * LEVELS + tid];
    __syncthreads();

    const float* xs = pre + (size_t)v * T_OBS * LEVELS;
    int cur = 0;
    for (int t = 0; t < TOTAL_T; ++t) {
        if (tid < LEVELS)
            s_u[tid] = (t < T_OBS) ? xs[(size_t)t * LEVELS + tid] : s_o[tid];
        __syncthreads();

        const int nxt = cur ^ 1;
        if (tid < NCELLS) {
            float s = s_b[tid];
            #pragma unroll
            for (int l = 0; l < LEVELS; ++l)
                s = fmaf(s_u[l], s_wx[l * NCELLS + tid], s);
            #pragma unroll 8
            for (int i = 0; i < NCELLS; ++i)
                s = fmaf(s_h[cur][i], s_wh[i * NCELLS + tid], s);
            const float f  = 1.0f / (1.0f + expf(-s));
            const float h  = s_h[cur][tid];
            const float hn = (h + DT_STEP * f * s_a[tid]) /
                             (1.0f + DT_STEP * (s_it[tid] + f));
            s_h[nxt][tid] = hn;
        }
        __syncthreads();

        if (tid < LEVELS) {
            float o = s_bout[tid];
            #pragma unroll 8
            for (int i = 0; i < NCELLS; ++i)
                o = fmaf(s_h[nxt][i], s_wout[i * LEVELS + tid], o);
            s_o[tid] = o;
            vs[(size_t)t * (VAR_N * LEVELS) + v * LEVELS + tid] = o;
        }
        cur = nxt;
    }
}

// ---------------------------------------------------------------------------
extern "C" void kernel_launch(void* const* d_in, const int* in_sizes, int n_in,
                              void* d_out, int out_size, void* d_ws, size_t ws_size,
                              hipStream_t stream)
{
    const float* x = (const float*)d_in[0];
    const float *pw[5], *pb[5], *c1w[5], *c1b[5], *c2w[5], *c2b[5];
    for (int i = 0; i < 5; ++i) { pw[i]  = (const float*)d_in[ 1 + 2*i]; pb[i]  = (const float*)d_in[ 2 + 2*i]; }
    for (int i = 0; i < 5; ++i) { c1w[i] = (const float*)d_in[11 + 2*i]; c1b[i] = (const float*)d_in[12 + 2*i]; }
    for (int i = 0; i < 5; ++i) { c2w[i] = (const float*)d_in[21 + 2*i]; c2b[i] = (const float*)d_in[22 + 2*i]; }
    const float* ltc_wx   = (const float*)d_in[31];
    const float* ltc_wh   = (const float*)d_in[32];
    const float* ltc_b    = (const float*)d_in[33];
    const float* ltc_a    = (const float*)d_in[34];
    const float* ltc_tau  = (const float*)d_in[35];
    const float* ltc_wout = (const float*)d_in[36];
    const float* ltc_bout = (const float*)d_in[37];
    float* out = (float*)d_out;

    // Workspace layout (floats).  BUF holds both 16432x68 and 4108x272.
    const size_t BUF = (size_t)(TOTAL_T * VAR_N) * DDIM;   // 1,117,376
    float* ws  = (float*)d_ws;
    float* b0  = ws;                                       // ping
    float* b1  = ws + BUF;                                 // pong
    float* pre = ws + 2 * BUF;                             // 4096*68
    float* vsb = pre + (size_t)T_OBS * DDIM;               // 4108*4*17
    float* c1o = vsb + (size_t)TOTAL_T * VAR_N * LEVELS;   // 16432*68 (== 4108x272)

    auto gemm = [&](const float* A, const float* W, const float* B, float* C,
                    int M, int N, int K, bool relu) {
        dim3 grid((M + 16 * MT - 1) / (16 * MT), (N + 15) / 16);
        if (relu)
            hipLaunchKernelGGL((wmma_gemm_bias<true>),  grid, dim3(32), 0, stream, A, W, B, C, M, N, K);
        else
            hipLaunchKernelGGL((wmma_gemm_bias<false>), grid, dim3(32), 0, stream, A, W, B, C, M, N, K);
    };

    // 1) preproc VarEncoder(68->68): 4 x (GEMM+ReLU), final GEMM (no ReLU)
    gemm(x,  pw[0], pb[0], b0,  T_OBS, DDIM, DDIM, true);
    gemm(b0, pw[1], pb[1], b1,  T_OBS, DDIM, DDIM, true);
    gemm(b1, pw[2], pb[2], b0,  T_OBS, DDIM, DDIM, true);
    gemm(b0, pw[3], pb[3], b1,  T_OBS, DDIM, DDIM, true);
    gemm(b1, pw[4], pb[4], pre, T_OBS, DDIM, DDIM, false);

    // 2) LTC scan: 4 vars, serial over 4108 steps, writes (4108,4,17)
    hipLaunchKernelGGL(ltc_scan, dim3(VAR_N), dim3(64), 0, stream,
                       pre, ltc_wx, ltc_wh, ltc_b, ltc_a, ltc_tau,
                       ltc_wout, ltc_bout, vsb);

    // 3) c1 VarEncoder(17->68) over 16432 rows, outer ReLU folded into last GEMM
    const int M1 = TOTAL_T * VAR_N;                        // 16432
    gemm(vsb, c1w[0], c1b[0], b0,  M1, LEVELS, LEVELS, true);
    gemm(b0,  c1w[1], c1b[1], b1,  M1, LEVELS, LEVELS, true);
    gemm(b1,  c1w[2], c1b[2], b0,  M1, DDIM,   LEVELS, true);
    gemm(b0,  c1w[3], c1b[3], b1,  M1, DDIM,   DDIM,   true);
    gemm(b1,  c1w[4], c1b[4], c1o, M1, DDIM,   DDIM,   true);  // outer relu

    // 4) c2 VarEncoder(272->68) over 4108 rows (c1o viewed as (4108,272))
    const int FLAT = VAR_N * DDIM;                         // 272
    gemm(c1o, c2w[0], c2b[0], b0,  TOTAL_T, FLAT, FLAT, true);
    gemm(b0,  c2w[1], c2b[1], b1,  TOTAL_T, FLAT, FLAT, true);
    gemm(b1,  c2w[2], c2b[2], b0,  TOTAL_T, DDIM, FLAT, true);
    gemm(b0,  c2w[3], c2b[3], b1,  TOTAL_T, DDIM, DDIM, true);
    gemm(b1,  c2w[4], c2b[4], out, TOTAL_T, DDIM, DDIM, false);
}